// GNNRegressor_67765993997190
// MI455X (gfx1250) — compile-verified
//
#include <hip/hip_runtime.h>
#include <cfloat>

typedef __attribute__((ext_vector_type(2))) float v2f;
typedef __attribute__((ext_vector_type(8))) float v8f;

#define TPB 256
static inline int cdiv_i(long long a, long long b) { return (int)((a + b - 1) / b); }

// ---------------- elementwise helpers ----------------
__global__ void k_fill(float* __restrict__ p, float v, size_t n) {
  size_t i = (size_t)blockIdx.x * blockDim.x + threadIdx.x;
  if (i < n) p[i] = v;
}

__global__ void k_fill_bias(float* __restrict__ p, const float* __restrict__ bias,
                            size_t n, int F) {
  size_t i = (size_t)blockIdx.x * blockDim.x + threadIdx.x;
  if (i < n) p[i] = bias[i % F];
}

__global__ void k_relu(float* __restrict__ p, size_t n) {
  size_t i = (size_t)blockIdx.x * blockDim.x + threadIdx.x;
  if (i < n) p[i] = fmaxf(p[i], 0.0f);
}

__global__ void k_rsqrt(float* __restrict__ p, int n) {
  int i = blockIdx.x * blockDim.x + threadIdx.x;
  if (i < n) p[i] = rsqrtf(p[i]);
}

// ---------------- degree ----------------
__global__ void k_deg(const int* __restrict__ dst, float* __restrict__ deg, int E) {
  int i = blockIdx.x * blockDim.x + threadIdx.x;
  if (i < E) unsafeAtomicAdd(&deg[dst[i]], 1.0f);
}

// ---------------- propagation (gather-scale-scatter) ----------------
// 16-feature prop: one thread per (edge, feature). Edges [E, E+N) are self-loops.
__global__ void k_prop16(const float* __restrict__ h, const int* __restrict__ src,
                         const int* __restrict__ dst, const float* __restrict__ dis,
                         float* __restrict__ out, int E, int N) {
  int t = blockIdx.x * blockDim.x + threadIdx.x;
  int e = t >> 4, f = t & 15;
  if (e >= E + N) return;
  int s, d;
  if (e < E) { s = src[e]; d = dst[e]; } else { s = d = e - E; }
  float w = dis[s] * dis[d];
  unsafeAtomicAdd(&out[(size_t)d * 16 + f], h[(size_t)s * 16 + f] * w);
}

// 64-feature prop: one wave per edge, each lane handles 2 features (coalesced).
__global__ void k_prop64(const float* __restrict__ h, const int* __restrict__ src,
                         const int* __restrict__ dst, const float* __restrict__ dis,
                         float* __restrict__ out, int E, int N) {
  int wave = (blockIdx.x * blockDim.x + threadIdx.x) >> 5;
  int lane = threadIdx.x & 31;
  if (wave >= E + N) return;
  int s, d;
  if (wave < E) { s = src[wave]; d = dst[wave]; } else { s = d = wave - E; }
  float w = dis[s] * dis[d];
  size_t sb = (size_t)s * 64, db = (size_t)d * 64;
  float v0 = h[sb + lane] * w;
  float v1 = h[sb + lane + 32] * w;
  unsafeAtomicAdd(&out[db + lane], v0);
  unsafeAtomicAdd(&out[db + lane + 32], v1);
}

// ---------------- f32 WMMA GEMM: C[M,Nout] = A[M,K] @ W[K,Nout] (+bias)(+relu) ----
// One wave computes one 16x16 tile via V_WMMA_F32_16X16X4_F32, stepping K by 4.
// A-matrix lane layout (ISA 7.12.2): lane m=lane&15, koff=2*(lane>>4):
//   vgpr0 = A[m][k+koff], vgpr1 = A[m][k+koff+1].   B mirrored with N in lane&15.
// C/D: acc[r] = C[r + 8*(lane>>4)][lane&15].
template <bool BIAS, bool RELU>
__global__ __launch_bounds__(TPB)
void k_gemm_wmma(const float* __restrict__ A, const float* __restrict__ W,
                 const float* __restrict__ bias, float* __restrict__ C,
                 int K, int Nout, int nTiles) {
  int wave = (blockIdx.x * blockDim.x + threadIdx.x) >> 5;
  if (wave >= nTiles) return;                 // wave-uniform; EXEC stays all-ones
  int lane  = threadIdx.x & 31;
  int tilesN = Nout >> 4;
  int tm = wave / tilesN;
  int tn = wave % tilesN;
  int half  = lane >> 4;                      // 0 or 1
  int idx16 = lane & 15;
  int koff  = half << 1;

  const float* Arow = A + (size_t)(tm * 16 + idx16) * K;
  const float* Wcol = W + (size_t)(tn * 16 + idx16);

  v8f acc = {};
  for (int k = 0; k < K; k += 4) {
    v2f a, b;
    a.x = Arow[k + koff];
    a.y = Arow[k + koff + 1];
    b.x = Wcol[(size_t)(k + koff) * Nout];
    b.y = Wcol[(size_t)(k + koff + 1) * Nout];
    acc = __builtin_amdgcn_wmma_f32_16x16x4_f32(false, a, false, b,
                                                (short)0, acc, false, false);
  }

  int col = tn * 16 + idx16;
  float bv = BIAS ? bias[col] : 0.0f;
#pragma unroll
  for (int r = 0; r < 8; ++r) {
    int mrow = tm * 16 + r + (half << 3);
    float v = acc[r] + bv;
    if (RELU) v = fmaxf(v, 0.0f);
    C[(size_t)mrow * Nout + col] = v;
  }
}

// ---------------- segment max (batch is sorted) ----------------
__device__ __forceinline__ void atomicMaxF(float* addr, float val) {
  int* ia = (int*)addr;
  int old = __float_as_int(*addr);
  while (__int_as_float(old) < val) {
    int assumed = old;
    old = atomicCAS(ia, assumed, __float_as_int(val));
    if (old == assumed) break;
  }
}

__global__ void k_segmax(const float* __restrict__ z, const int* __restrict__ batch,
                         float* __restrict__ g, int N) {
  int f = blockIdx.x;           // feature 0..63
  int t = threadIdx.x;          // 0..TPB-1
  float curMax = -FLT_MAX;
  int curG = -1;
  for (int i = t; i < N; i += TPB) {
    int b = batch[i];
    float v = z[(size_t)i * 64 + f];
    if (b != curG) {
      if (curG >= 0) atomicMaxF(&g[curG * 64 + f], curMax);
      curG = b;
      curMax = v;
    } else {
      curMax = fmaxf(curMax, v);
    }
  }
  if (curG >= 0) atomicMaxF(&g[curG * 64 + f], curMax);
}

// ---------------- tiny head MLP: 8 graphs, one lane each ----------------
__global__ void k_head(const float* __restrict__ g,
                       const float* __restrict__ Wf1, const float* __restrict__ bf1,
                       const float* __restrict__ Wf2, const float* __restrict__ bf2,
                       const float* __restrict__ Wc,  const float* __restrict__ bc,
                       const float* __restrict__ Wcb, const float* __restrict__ bcb,
                       float* __restrict__ out) {
  int gi = threadIdx.x;
  if (gi >= 8) return;
  const float* gr = g + gi * 64;
  float h1[32];
#pragma unroll
  for (int j = 0; j < 32; ++j) {
    float acc = bf1[j];
    for (int k = 0; k < 64; ++k) acc += gr[k] * Wf1[k * 32 + j];
    h1[j] = fmaxf(acc, 0.0f);
  }
  float h2[16];
#pragma unroll
  for (int j = 0; j < 16; ++j) {
    float acc = bf2[j];
#pragma unroll
    for (int k = 0; k < 32; ++k) acc += h1[k] * Wf2[k * 16 + j];
    h2[j] = fmaxf(acc, 0.0f);
  }
  float c = bc[0], cb = bcb[0];
#pragma unroll
  for (int k = 0; k < 16; ++k) { c += h2[k] * Wc[k]; cb += h2[k] * Wcb[k]; }
  out[gi] = c;
  out[8 + gi] = cb;
}

// ---------------- launcher ----------------
extern "C" void kernel_launch(void* const* d_in, const int* in_sizes, int n_in,
                              void* d_out, int out_size, void* d_ws, size_t ws_size,
                              hipStream_t stream) {
  const float* x     = (const float*)d_in[0];
  const int*   ei    = (const int*)d_in[1];
  const int*   batch = (const int*)d_in[2];
  const float* W1  = (const float*)d_in[3];  const float* b1  = (const float*)d_in[4];
  const float* W2  = (const float*)d_in[5];  const float* b2  = (const float*)d_in[6];
  const float* Wg1 = (const float*)d_in[7];  const float* bg1 = (const float*)d_in[8];
  const float* Wg2 = (const float*)d_in[9];  const float* bg2 = (const float*)d_in[10];
  const float* Wg3 = (const float*)d_in[11]; const float* bg3 = (const float*)d_in[12];
  const float* Wsg = (const float*)d_in[13]; const float* bsg = (const float*)d_in[14];
  const float* Wf1 = (const float*)d_in[15]; const float* bf1 = (const float*)d_in[16];
  const float* Wf2 = (const float*)d_in[17]; const float* bf2 = (const float*)d_in[18];
  const float* Wc  = (const float*)d_in[19]; const float* bc  = (const float*)d_in[20];
  const float* Wcb = (const float*)d_in[21]; const float* bcb = (const float*)d_in[22];
  float* out = (float*)d_out;

  const int N = in_sizes[0] / 16;   // 100000 (divisible by 16: 6250 tiles)
  const int E = in_sizes[1] / 2;    // 1200000
  const int* src = ei;
  const int* dst = ei + E;

  // workspace carve-up
  float* dis  = (float*)d_ws;                      // N
  float* bufA = dis + N;                           // N*128
  float* bufB = bufA + (size_t)N * 128;            // N*64
  float* bufC = bufB + (size_t)N * 64;             // N*64
  float* g    = bufC + (size_t)N * 64;             // 8*64

  const size_t n64 = (size_t)N * 64;
  const int waves64 = E + N;                                 // one wave per edge
  const int gridProp64 = cdiv_i((long long)waves64 * 32, TPB);
  const int mt = N / 16;                                     // 6250 row tiles

  // degree -> dis = rsqrt(deg) (self-loops give deg >= 1)
  k_fill<<<cdiv_i(N, TPB), TPB, 0, stream>>>(dis, 1.0f, (size_t)N);
  k_deg<<<cdiv_i(E, TPB), TPB, 0, stream>>>(dst, dis, E);
  k_rsqrt<<<cdiv_i(N, TPB), TPB, 0, stream>>>(dis, N);

  // layer 1: z1 = relu(prop(x) @ W1 + b1)   [prop at 16 dims, then 16->128 GEMM]
  k_fill<<<cdiv_i((long long)N * 16, TPB), TPB, 0, stream>>>(bufB, 0.0f, (size_t)N * 16);
  k_prop16<<<cdiv_i((long long)(E + N) * 16, TPB), TPB, 0, stream>>>(x, src, dst, dis, bufB, E, N);
  {
    int nt = mt * (128 / 16);
    k_gemm_wmma<true, true><<<cdiv_i((long long)nt * 32, TPB), TPB, 0, stream>>>(
        bufB, W1, b1, bufA, 16, 128, nt);
  }

  // layer 2: z2 = prop(z1 @ W2) + b2        [128->64 GEMM first, prop at 64]
  {
    int nt = mt * 4;
    k_gemm_wmma<false, false><<<cdiv_i((long long)nt * 32, TPB), TPB, 0, stream>>>(
        bufA, W2, nullptr, bufB, 128, 64, nt);
  }
  k_fill_bias<<<cdiv_i((long long)n64, TPB), TPB, 0, stream>>>(bufC, b2, n64, 64);
  k_prop64<<<gridProp64, TPB, 0, stream>>>(bufB, src, dst, dis, bufC, E, N);

  // stacked GCN layers: z = relu(prop(z @ Wg) + bg)
  const float* Wg[3] = {Wg1, Wg2, Wg3};
  const float* bg[3] = {bg1, bg2, bg3};
  float* in3[3]  = {bufC, bufA, bufC};
  float* out3[3] = {bufA, bufC, bufA};
  for (int l = 0; l < 3; ++l) {
    int nt = mt * 4;
    k_gemm_wmma<false, false><<<cdiv_i((long long)nt * 32, TPB), TPB, 0, stream>>>(
        in3[l], Wg[l], nullptr, bufB, 64, 64, nt);
    k_fill_bias<<<cdiv_i((long long)n64, TPB), TPB, 0, stream>>>(out3[l], bg[l], n64, 64);
    k_prop64<<<gridProp64, TPB, 0, stream>>>(bufB, src, dst, dis, out3[l], E, N);
    k_relu<<<cdiv_i((long long)n64, TPB), TPB, 0, stream>>>(out3[l], n64);
  }
  // z5 now in bufA

  // SGConv: h = P^4 z5 ; z = h @ Wsg + bsg
  float* pp[5] = {bufA, bufC, bufA, bufC, bufA};  // ping-pong, ends in bufA
  for (int p = 0; p < 4; ++p) {
    k_fill<<<cdiv_i((long long)n64, TPB), TPB, 0, stream>>>(pp[p + 1], 0.0f, n64);
    k_prop64<<<gridProp64, TPB, 0, stream>>>(pp[p], src, dst, dis, pp[p + 1], E, N);
  }
  {
    int nt = mt * 4;
    k_gemm_wmma<true, false><<<cdiv_i((long long)nt * 32, TPB), TPB, 0, stream>>>(
        bufA, Wsg, bsg, bufB, 64, 64, nt);
  }

  // global max pool per graph + head MLP
  k_fill<<<cdiv_i(8 * 64, TPB), TPB, 0, stream>>>(g, -FLT_MAX, (size_t)(8 * 64));
  k_segmax<<<64, TPB, 0, stream>>>(bufB, batch, g, N);
  k_head<<<1, 32, 0, stream>>>(g, Wf1, bf1, Wf2, bf2, Wc, bc, Wcb, bcb, out);
}